// ReconstructionLoss_59803124629910
// MI455X (gfx1250) — compile-verified
//
#include <hip/hip_runtime.h>
#include <hip/hip_bf16.h>
#include <math.h>

// ---------------------------------------------------------------------------
// ReconstructionLoss for MI455X (gfx1250, wave32).
// Bandwidth-bound fused affine-warp + L1 + masked mean reduction.
//  - img_R streamed via global_load_b128 (4 consecutive pixels per thread)
//  - fully-OOB pixel groups skip all memory traffic
//  - cross-lane reductions via V_WMMA_F32_16X16X4_F32 with all-ones B
//    (exact fp32 row-sum reduction on the CDNA5 matrix pipe)
// ---------------------------------------------------------------------------

typedef __attribute__((ext_vector_type(2))) float v2f;
typedef __attribute__((ext_vector_type(8))) float v8f;

#define WW     512
#define HHH    512
#define PLANE  (WW * HHH)             // 262144
#define CH     3
#define NBATCH 32
#define TPB    256
#define PPT    8                      // pixels per thread (2 groups of 4)
#define NBLK   (PLANE / (TPB * PPT))  // 128 blocks per image

// Exact fp32 sum of one value per lane across a wave32, via WMMA.
// A (16x4 f32): lane L<16 holds A(L,0)=v, lane L>=16 holds A(L-16,2)=v,
// K=1,3 are zero. B = all ones -> D(m,n) = v_m + v_{m+16}.
// Per-lane sum of the 8 D VGPRs gives rows 0..7 (lanes 0-15) or rows 8..15
// (lanes 16-31); one xor-16 shuffle completes the 32-lane sum on all lanes.
__device__ __forceinline__ float wave_sum32(float v) {
    v2f a;  a.x = v;     a.y = 0.0f;
    v2f bo; bo.x = 1.0f; bo.y = 1.0f;
    v8f c = {};
    v8f d = __builtin_amdgcn_wmma_f32_16x16x4_f32(
        /*neg_a=*/false, a, /*neg_b=*/false, bo,
        /*c_mod=*/(short)0, c, /*reuse_a=*/false, /*reuse_b=*/false);
    float s = ((d[0] + d[1]) + (d[2] + d[3])) + ((d[4] + d[5]) + (d[6] + d[7]));
    s += __shfl_xor(s, 16, 32);
    return s;
}

__global__ __launch_bounds__(TPB)
void warp_loss_partial(const float* __restrict__ theta,
                       const float* __restrict__ imgR,
                       const float* __restrict__ imgO,
                       float* __restrict__ numPart,
                       float* __restrict__ denPart) {
    const int b = blockIdx.y;

    // Per-batch affine (cx=cy=256, sw=sh=511 so sh/sw == 1).
    const float ang = theta[b * 3 + 0];
    const float scl = theta[b * 3 + 1];
    const float shf = 100.0f * theta[b * 3 + 2];
    float sn, cs;
    sincosf(ang, &sn, &cs);
    const float alpha = scl * cs;
    const float beta  = scl * sn;
    const float tx = (1.0f - alpha) * 256.0f - beta * 256.0f + shf;
    const float ty = beta * 256.0f + (1.0f - alpha) * 256.0f;
    const float Am = alpha, Bm = beta;
    const float Cm = alpha + beta + 2.0f * tx * (1.0f / 511.0f) - 1.0f;
    const float Dm = -beta, Em = alpha;
    const float Fm = -beta + alpha + 2.0f * ty * (1.0f / 511.0f) - 1.0f;

    const float* rB = imgR + (size_t)b * (CH * PLANE);
    const float* oB = imgO + (size_t)b * (CH * PLANE);

    float num = 0.0f, den = 0.0f;
    // Thread handles 2 groups of 4 consecutive pixels (16B aligned; a
    // 4-aligned group never crosses a row since 512 % 4 == 0).
    const int p0 = blockIdx.x * (TPB * PPT) + threadIdx.x * 4;

#pragma unroll
    for (int g = 0; g < PPT / 4; ++g) {
        const int p = p0 + g * (TPB * 4);
        const int h = p >> 9;
        const int wbase = p & (WW - 1);
        const float v = fmaf((float)h, 2.0f / 511.0f, -1.0f);
        const float gy_base = Em * v + Fm;
        const float gx_base = Bm * v + Cm;

        // ---- pass 1: pure VALU — sample coords + mask weights ----
        float xs[4], ys[4], wmv[4];
#pragma unroll
        for (int j = 0; j < 4; ++j) {
            const float u = fmaf((float)(wbase + j), 2.0f / 511.0f, -1.0f);
            const float x = (Am * u + gx_base + 1.0f) * 255.5f;
            const float y = (Dm * u + gy_base + 1.0f) * 255.5f;
            const float x0 = floorf(x), y0 = floorf(y);
            const float x1 = x0 + 1.0f, y1 = y0 + 1.0f;
            const float wx1 = x - x0, wy1 = y - y0;
            const float wx0 = 1.0f - wx1, wy0 = 1.0f - wy1;
            const bool vx0 = (x0 >= 0.0f) && (x0 < 512.0f);
            const bool vx1 = (x1 >= 0.0f) && (x1 < 512.0f);
            const bool vy0 = (y0 >= 0.0f) && (y0 < 512.0f);
            const bool vy1 = (y1 >= 0.0f) && (y1 < 512.0f);
            const float w00 = (vx0 && vy0) ? wx0 * wy0 : 0.0f;
            const float w10 = (vx1 && vy0) ? wx1 * wy0 : 0.0f;
            const float w01 = (vx0 && vy1) ? wx0 * wy1 : 0.0f;
            const float w11 = (vx1 && vy1) ? wx1 * wy1 : 0.0f;
            const float wm = (w00 + w10) + (w01 + w11);
            xs[j] = x; ys[j] = y; wmv[j] = wm;
            den += wm;
        }

        // ---- pass 2: memory traffic only if any pixel lands in-bounds ----
        if (((wmv[0] + wmv[1]) + (wmv[2] + wmv[3])) > 0.0f) {
            // Streaming reads: one B128 per channel for 4 pixels.
            const float4 r0 = *(const float4*)(rB + 0 * PLANE + p);
            const float4 r1 = *(const float4*)(rB + 1 * PLANE + p);
            const float4 r2 = *(const float4*)(rB + 2 * PLANE + p);
            const float rr0[4] = {r0.x, r0.y, r0.z, r0.w};
            const float rr1[4] = {r1.x, r1.y, r1.z, r1.w};
            const float rr2[4] = {r2.x, r2.y, r2.z, r2.w};

#pragma unroll
            for (int j = 0; j < 4; ++j) {
                if (wmv[j] > 0.0f) {
                    const float x = xs[j], y = ys[j];
                    const float x0 = floorf(x), y0 = floorf(y);
                    const float x1 = x0 + 1.0f, y1 = y0 + 1.0f;
                    const float wx1 = x - x0, wy1 = y - y0;
                    const float wx0 = 1.0f - wx1, wy0 = 1.0f - wy1;
                    const bool vx0 = (x0 >= 0.0f) && (x0 < 512.0f);
                    const bool vx1 = (x1 >= 0.0f) && (x1 < 512.0f);
                    const bool vy0 = (y0 >= 0.0f) && (y0 < 512.0f);
                    const bool vy1 = (y1 >= 0.0f) && (y1 < 512.0f);
                    const float w00 = (vx0 && vy0) ? wx0 * wy0 : 0.0f;
                    const float w10 = (vx1 && vy0) ? wx1 * wy0 : 0.0f;
                    const float w01 = (vx0 && vy1) ? wx0 * wy1 : 0.0f;
                    const float w11 = (vx1 && vy1) ? wx1 * wy1 : 0.0f;

                    const int ix0 = (int)fminf(fmaxf(x0, 0.0f), 511.0f);
                    const int ix1 = (int)fminf(fmaxf(x1, 0.0f), 511.0f);
                    const int iy0 = (int)fminf(fmaxf(y0, 0.0f), 511.0f);
                    const int iy1 = (int)fminf(fmaxf(y1, 0.0f), 511.0f);
                    const int o00 = iy0 * WW + ix0, o10 = iy0 * WW + ix1;
                    const int o01 = iy1 * WW + ix0, o11 = iy1 * WW + ix1;

                    const float* oc0 = oB;
                    const float* oc1 = oB + PLANE;
                    const float* oc2 = oB + 2 * PLANE;
                    const float wp0 = w00 * oc0[o00] + w10 * oc0[o10] +
                                      w01 * oc0[o01] + w11 * oc0[o11];
                    const float wp1 = w00 * oc1[o00] + w10 * oc1[o10] +
                                      w01 * oc1[o01] + w11 * oc1[o11];
                    const float wp2 = w00 * oc2[o00] + w10 * oc2[o10] +
                                      w01 * oc2[o01] + w11 * oc2[o11];
                    const float nrm = fabsf(rr0[j] - wp0) +
                                      fabsf(rr1[j] - wp1) +
                                      fabsf(rr2[j] - wp2);
                    num += wmv[j] * nrm;
                }
            }
        }
    }

    // Convergent from here: WMMA wave reduction (EXEC all ones).
    const float nw = wave_sum32(num);
    const float dw = wave_sum32(den);

    __shared__ float sN[TPB / 32];
    __shared__ float sD[TPB / 32];
    const int wv = threadIdx.x >> 5;
    if ((threadIdx.x & 31) == 0) { sN[wv] = nw; sD[wv] = dw; }
    __syncthreads();
    if (threadIdx.x == 0) {
        float n = 0.0f, d = 0.0f;
#pragma unroll
        for (int k = 0; k < TPB / 32; ++k) { n += sN[k]; d += sD[k]; }
        numPart[b * NBLK + blockIdx.x] = n;
        denPart[b * NBLK + blockIdx.x] = d;
    }
}

__global__ __launch_bounds__(1024)
void finalize_loss(const float* __restrict__ numPart,
                   const float* __restrict__ denPart,
                   float* __restrict__ out) {
    // 32 waves, wave b reduces batch b's 128 partials in a fixed order.
    const int b = threadIdx.x >> 5;
    const int lane = threadIdx.x & 31;

    float n = 0.0f, d = 0.0f;
#pragma unroll
    for (int i = 0; i < NBLK / 32; ++i) {
        const int idx = b * NBLK + i * 32 + lane;
        n += numPart[idx];
        d += denPart[idx];
    }
    n = wave_sum32(n);
    d = wave_sum32(d);

    __shared__ float sL[NBATCH];
    if (lane == 0) sL[b] = 10.0f * (n + 1e-16f) / (d + 1e-16f);
    __syncthreads();
    if (threadIdx.x == 0) {
        float s = 0.0f;
#pragma unroll
        for (int i = 0; i < NBATCH; ++i) s += sL[i];
        out[0] = s * (1.0f / 32.0f);
    }
}

extern "C" void kernel_launch(void* const* d_in, const int* in_sizes, int n_in,
                              void* d_out, int out_size, void* d_ws, size_t ws_size,
                              hipStream_t stream) {
    const float* theta = (const float*)d_in[0];
    const float* imgR  = (const float*)d_in[1];
    const float* imgO  = (const float*)d_in[2];
    float* out = (float*)d_out;

    // Workspace: 2 * 32 * 128 floats = 32 KB of deterministic partials.
    float* numPart = (float*)d_ws;
    float* denPart = numPart + NBATCH * NBLK;

    dim3 grid(NBLK, NBATCH);
    warp_loss_partial<<<grid, TPB, 0, stream>>>(theta, imgR, imgO, numPart, denPart);
    finalize_loss<<<1, 1024, 0, stream>>>(numPart, denPart, out);

    (void)in_sizes; (void)n_in; (void)out_size; (void)ws_size;
}